// NotAllPatch_22694607192478
// MI455X (gfx1250) — compile-verified
//
#include <hip/hip_runtime.h>
#include <math.h>

typedef __attribute__((ext_vector_type(2))) float v2f;
typedef __attribute__((ext_vector_type(8))) float v8f;

#define B_   64
#define S_   576
#define DIM_ 768
#define HID_ 128
#define KTOP 63               // KEEP_TOKEN - 1
#define NROW (B_ * 64 + B_)   // 4160 rows: visual tokens then cls rows

// ---------------------------------------------------------------------------
// Kernel 1: cls attention scores.  attn[b,s] = dot(cls_emb[b], fm[b,s])
// One block per batch, 8 waves; wave handles token s; float4 (B128) loads.
// ---------------------------------------------------------------------------
__global__ void attn_kernel(const float* __restrict__ cls_emb,
                            const float* __restrict__ fm,
                            float* __restrict__ attn) {
    __shared__ __align__(16) float cls_s[DIM_];
    const int b = blockIdx.x;

    const float4* cls4 = (const float4*)(cls_emb + (size_t)b * DIM_);
    for (int i = threadIdx.x; i < DIM_ / 4; i += 256)
        ((float4*)cls_s)[i] = cls4[i];
    __syncthreads();

    const int wave = threadIdx.x >> 5;
    const int lane = threadIdx.x & 31;
    for (int s = wave; s < S_; s += 8) {
        const float4* row4 = (const float4*)(fm + ((size_t)b * S_ + s) * DIM_);
        float sum = 0.f;
        for (int i = lane; i < DIM_ / 4; i += 32) {   // 6 iterations of B128
            float4 r = row4[i];
            float4 c = ((const float4*)cls_s)[i];
            sum += r.x * c.x + r.y * c.y + r.z * c.z + r.w * c.w;
        }
        for (int off = 16; off > 0; off >>= 1)
            sum += __shfl_xor(sum, off, 32);
        if (lane == 0) attn[b * S_ + s] = sum;
    }
}

// ---------------------------------------------------------------------------
// Kernel 2: per-batch top-63 (descending, min-index ties, like lax.top_k),
// masked softmax over all S logits, pooled extra token.
// One block (256 threads) per batch.
// ---------------------------------------------------------------------------
__global__ void topk_extra_kernel(const float* __restrict__ fm,
                                  const float* __restrict__ attn,
                                  int* __restrict__ idx_ws,
                                  float* __restrict__ extra_ws) {
    __shared__ float val[S_];
    __shared__ float work[S_];
    __shared__ float wgt[S_];
    __shared__ float rv[256];
    __shared__ int   ri[256];

    const int b = blockIdx.x;
    const int tid = threadIdx.x;

    for (int s = tid; s < S_; s += 256) {
        float v = attn[b * S_ + s];
        val[s] = v;
        work[s] = v;
    }
    __syncthreads();

    // 63 rounds of argmax (value desc, index asc on ties)
    for (int k = 0; k < KTOP; ++k) {
        float best = -INFINITY;
        int bi = S_;
        for (int s = tid; s < S_; s += 256) {
            float v = work[s];
            if (v > best || (v == best && s < bi)) { best = v; bi = s; }
        }
        rv[tid] = best; ri[tid] = bi;
        __syncthreads();
        for (int off = 128; off > 0; off >>= 1) {
            if (tid < off) {
                float v2 = rv[tid + off]; int i2 = ri[tid + off];
                if (v2 > rv[tid] || (v2 == rv[tid] && i2 < ri[tid])) {
                    rv[tid] = v2; ri[tid] = i2;
                }
            }
            __syncthreads();
        }
        const int sel = ri[0];
        if (tid == 0) {
            idx_ws[b * 64 + k] = sel;
            work[sel] = -INFINITY;   // mark selected
        }
        __syncthreads();
    }

    // masked logits: selected positions get -10000
    float m = -INFINITY;
    for (int s = tid; s < S_; s += 256) {
        float x = val[s] + (work[s] == -INFINITY ? -10000.0f : 0.0f);
        wgt[s] = x;
        m = fmaxf(m, x);
    }
    rv[tid] = m;
    __syncthreads();
    for (int off = 128; off > 0; off >>= 1) {
        if (tid < off) rv[tid] = fmaxf(rv[tid], rv[tid + off]);
        __syncthreads();
    }
    m = rv[0];
    __syncthreads();

    float lsum = 0.f;
    for (int s = tid; s < S_; s += 256) {
        float e = expf(wgt[s] - m);
        wgt[s] = e;
        lsum += e;
    }
    rv[tid] = lsum;
    __syncthreads();
    for (int off = 128; off > 0; off >>= 1) {
        if (tid < off) rv[tid] += rv[tid + off];
        __syncthreads();
    }
    const float inv = 1.0f / rv[0];
    __syncthreads();

    // extra[b, d4*4..d4*4+3] = inv * sum_s wgt[s] * fm[b,s,d4]  (B128 loads)
    for (int d4 = tid; d4 < DIM_ / 4; d4 += 256) {   // 192 active threads
        const float4* base = (const float4*)(fm + (size_t)b * S_ * DIM_) + d4;
        float ax = 0.f, ay = 0.f, az = 0.f, aw = 0.f;
        for (int s = 0; s < S_; ++s) {
            float4 v = base[(size_t)s * (DIM_ / 4)];
            float w = wgt[s];
            ax += w * v.x; ay += w * v.y; az += w * v.z; aw += w * v.w;
        }
        float4 o; o.x = ax * inv; o.y = ay * inv; o.z = az * inv; o.w = aw * inv;
        ((float4*)(extra_ws + (size_t)b * DIM_))[d4] = o;
    }
}

// ---------------------------------------------------------------------------
// Kernel 3: H = tanh(X @ W1 + b1), X rows gathered on the fly.
// One block per 16-row tile, 4 waves; each wave does 2 col-tiles of HID=128.
// fp32 WMMA 16x16x4.
// ---------------------------------------------------------------------------
#define LDA 776   // padded LDS row stride (floats); 776*4 bytes % 16 == 0

__global__ void gemm1_kernel(const float* __restrict__ fm,
                             const float* __restrict__ cls_emb,
                             const float* __restrict__ extra_ws,
                             const int* __restrict__ idx_ws,
                             const float* __restrict__ W1,
                             const float* __restrict__ b1,
                             float* __restrict__ H) {
    __shared__ __align__(16) float Xs[16 * LDA];
    __shared__ const float* rowptr[16];

    const int tile = blockIdx.x;
    const int tid = threadIdx.x;

    if (tid < 16) {
        const int r = tile * 16 + tid;
        const float* p;
        if (r < B_ * 64) {
            const int b = r >> 6, k = r & 63;
            p = (k < KTOP) ? fm + ((size_t)b * S_ + idx_ws[b * 64 + k]) * DIM_
                           : extra_ws + (size_t)b * DIM_;
        } else {
            p = cls_emb + (size_t)(r - B_ * 64) * DIM_;
        }
        rowptr[tid] = p;
    }
    __syncthreads();

    for (int i = tid; i < 16 * (DIM_ / 4); i += 128) {   // B128 staging
        const int rl = i / (DIM_ / 4);
        const int d4 = i - rl * (DIM_ / 4);
        ((float4*)(Xs + rl * LDA))[d4] = ((const float4*)rowptr[rl])[d4];
    }
    __syncthreads();

    const int wave  = tid >> 5;
    const int lane  = tid & 31;
    const int mrow  = lane & 15;          // A: M index / B: N index
    const int khalf = (lane >> 4) * 2;    // K pair base for this lane half

    for (int t = 0; t < 2; ++t) {
        const int col0 = (wave * 2 + t) * 16;
        v8f c = {};
        for (int k0 = 0; k0 < DIM_; k0 += 4) {
            v2f a, bb;
            a[0]  = Xs[mrow * LDA + k0 + khalf];
            a[1]  = Xs[mrow * LDA + k0 + khalf + 1];
            bb[0] = W1[(size_t)(k0 + khalf)     * HID_ + col0 + mrow];
            bb[1] = W1[(size_t)(k0 + khalf + 1) * HID_ + col0 + mrow];
            c = __builtin_amdgcn_wmma_f32_16x16x4_f32(
                    false, a, false, bb, (short)0, c, false, false);
        }
        const int n = col0 + mrow;
        const float bias = b1[n];
        const int rbase = tile * 16 + (lane >> 4) * 8;  // C/D: M = v + 8*(lane/16)
        #pragma unroll
        for (int v = 0; v < 8; ++v)
            H[(size_t)(rbase + v) * HID_ + n] = tanhf(c[v] + bias);
    }
}

// ---------------------------------------------------------------------------
// Kernel 4: Out = H @ W2 + b2, written directly into d_out
// (rows 0..4095 = ae(visual_tokens), rows 4096..4159 = ae(cls_emb)).
// One block per 16-row tile, 8 waves; each wave does 6 col-tiles of DIM=768.
// ---------------------------------------------------------------------------
#define LDH 132   // 132*4 bytes % 16 == 0

__global__ void gemm2_kernel(const float* __restrict__ H,
                             const float* __restrict__ W2,
                             const float* __restrict__ b2,
                             float* __restrict__ out) {
    __shared__ __align__(16) float Hs[16 * LDH];
    const int tile = blockIdx.x;
    const int tid = threadIdx.x;

    for (int i = tid; i < 16 * (HID_ / 4); i += 256) {   // B128 staging
        const int rl = i >> 5;            // / 32
        const int d4 = i & 31;
        ((float4*)(Hs + rl * LDH))[d4] =
            ((const float4*)(H + (size_t)(tile * 16 + rl) * HID_))[d4];
    }
    __syncthreads();

    const int wave  = tid >> 5;
    const int lane  = tid & 31;
    const int mrow  = lane & 15;
    const int khalf = (lane >> 4) * 2;

    for (int t = 0; t < 6; ++t) {
        const int col0 = (wave * 6 + t) * 16;
        v8f c = {};
        for (int k0 = 0; k0 < HID_; k0 += 4) {
            v2f a, bb;
            a[0]  = Hs[mrow * LDH + k0 + khalf];
            a[1]  = Hs[mrow * LDH + k0 + khalf + 1];
            bb[0] = W2[(size_t)(k0 + khalf)     * DIM_ + col0 + mrow];
            bb[1] = W2[(size_t)(k0 + khalf + 1) * DIM_ + col0 + mrow];
            c = __builtin_amdgcn_wmma_f32_16x16x4_f32(
                    false, a, false, bb, (short)0, c, false, false);
        }
        const int n = col0 + mrow;
        const float bias = b2[n];
        const int rbase = tile * 16 + (lane >> 4) * 8;
        #pragma unroll
        for (int v = 0; v < 8; ++v)
            out[(size_t)(rbase + v) * DIM_ + n] = c[v] + bias;
    }
}

// ---------------------------------------------------------------------------
extern "C" void kernel_launch(void* const* d_in, const int* in_sizes, int n_in,
                              void* d_out, int out_size, void* d_ws, size_t ws_size,
                              hipStream_t stream) {
    const float* cls_emb = (const float*)d_in[0];   // [B, DIM]
    const float* fm      = (const float*)d_in[1];   // [B, S, DIM]
    const float* W1      = (const float*)d_in[2];   // [DIM, HID]
    const float* b1      = (const float*)d_in[3];   // [HID]
    const float* W2      = (const float*)d_in[4];   // [HID, DIM]
    const float* b2      = (const float*)d_in[5];   // [DIM]
    float* out = (float*)d_out;                     // [NROW, DIM] flat

    // Workspace layout (floats):
    //   [0, 36864)            attn        B*S
    //   [36864, 40960)        idx (int)   B*64
    //   [40960, 90112)        extra       B*DIM
    //   [90112, 622592)       H           NROW*HID
    float* ws      = (float*)d_ws;
    float* attn    = ws;
    int*   idx_ws  = (int*)(ws + (size_t)B_ * S_);
    float* extra   = ws + (size_t)B_ * S_ + (size_t)B_ * 64;
    float* H       = extra + (size_t)B_ * DIM_;

    attn_kernel<<<B_, 256, 0, stream>>>(cls_emb, fm, attn);
    topk_extra_kernel<<<B_, 256, 0, stream>>>(fm, attn, idx_ws, extra);
    gemm1_kernel<<<NROW / 16, 128, 0, stream>>>(fm, cls_emb, extra, idx_ws, W1, b1, H);
    gemm2_kernel<<<NROW / 16, 256, 0, stream>>>(H, W2, b2, out);
}